// EGNNLayer_45500883534041
// MI455X (gfx1250) — compile-verified
//
#include <hip/hip_runtime.h>
#include <hip/hip_bf16.h>

// ---------------------------------------------------------------------------
// E(3)-equivariant GNN layer for MI455X (gfx1250, wave32).
// Dense contractions -> V_WMMA_F32_16X16X4_F32.
// Weight streaming -> GLOBAL_LOAD_ASYNC_TO_LDS_B128 (double buffered),
// synchronized with s_wait_asynccnt.
// ---------------------------------------------------------------------------

typedef __attribute__((ext_vector_type(2))) float v2f;
typedef __attribute__((ext_vector_type(8))) float v8f;
typedef __attribute__((ext_vector_type(4))) int   v4i;

static __device__ __forceinline__ v8f wmma_f32x4(v2f a, v2f b, v8f c) {
  // D = A(16x4) x B(4x16) + C(16x16), fp32
  return __builtin_amdgcn_wmma_f32_16x16x4_f32(false, a, false, b,
                                               (short)0, c, false, false);
}

#if __has_builtin(__builtin_amdgcn_global_load_async_to_lds_b128)
#define HAS_ASYNC_LDS 1
#else
#define HAS_ASYNC_LDS 0
#endif

typedef __attribute__((address_space(1))) v4i g1_v4i;  // global (AS1) int4
typedef __attribute__((address_space(3))) v4i l3_v4i;  // LDS (AS3) int4

// Copy 16 bytes global -> LDS (per active lane). Async on CDNA5.
static __device__ __forceinline__ void async_copy16(const float* gsrc,
                                                    float* ldst) {
#if HAS_ASYNC_LDS
  __builtin_amdgcn_global_load_async_to_lds_b128((g1_v4i*)gsrc,
                                                 (l3_v4i*)ldst, 0, 0);
#else
  *(float4*)ldst = *(const float4*)gsrc;
#endif
}

static __device__ __forceinline__ void async_wait0() {
#if HAS_ASYNC_LDS
  asm volatile("s_wait_asynccnt 0" ::: "memory");
#endif
}

#define S96       0.10206207261596575f   // 1/sqrt(96)
#define INVSQRT3  0.57735026918962576f
#define INVSQRT32 0.17677669529663687f   // 1/sqrt(32)
#define S5120     0.013975424859373686f  // 1/sqrt(5120)
#define S4096     0.015625f              // 1/sqrt(4096)

#define V8ZERO {0.f,0.f,0.f,0.f,0.f,0.f,0.f,0.f}

// ---------------------------------------------------------------------------
// K0: assemble Wbig (160 x 544) for the per-node message factors, and
//     Wperm (64 x 1024): Wperm[v*1024 + u*32 + w] = wu101[u,v,w].
// F-row layout per node (544 floats):
//   [0:64)    A0s[w]     = sum_u x0[u] wm000[u,0,w] * S96
//   [64:256)  T110s[w,k] = sum_u x1[u,k] wm110[u,0,w] * S96/sqrt3   (w*3+k)
//   [256:288) C011s[w]   = sum_u x0[u] wm011[u,0,w] * S96
//   [288:384) C101s[w,k] = sum_u x1[u,k] wm101[u,0,w] * S96
//   [384:448) D0[w]      = (x0 @ lm0)/8
//   [448:544) D1[w,k]    = sum_u x1[u,k] lm1[u,w] / sqrt(32)
// ---------------------------------------------------------------------------
__global__ void k_build_weights(const float* __restrict__ wm000,
                                const float* __restrict__ wm011,
                                const float* __restrict__ wm101,
                                const float* __restrict__ wm110,
                                const float* __restrict__ lm0,
                                const float* __restrict__ lm1,
                                const float* __restrict__ wu101,
                                float* __restrict__ Wbig,
                                float* __restrict__ Wperm) {
  int idx = blockIdx.x * blockDim.x + threadIdx.x;
  const int NB = 160 * 544;
  if (idx < NB) {
    int i = idx / 544, j = idx - i * 544;
    float val = 0.f;
    if (i < 64) {
      int u = i;
      if (j < 64)                        val = wm000[u * 64 + j] * S96;
      else if (j >= 256 && j < 288)      val = wm011[u * 32 + (j - 256)] * S96;
      else if (j >= 384 && j < 448)      val = lm0[u * 64 + (j - 384)] * 0.125f;
    } else {
      int t = i - 64, u = t / 3, k = t - 3 * (t / 3);
      if (j >= 64 && j < 256) {
        int s = j - 64, w = s / 3, k2 = s - 3 * w;
        if (k == k2) val = wm110[u * 64 + w] * (S96 * INVSQRT3);
      } else if (j >= 288 && j < 384) {
        int s = j - 288, w = s / 3, k2 = s - 3 * w;
        if (k == k2) val = wm101[u * 32 + w] * S96;
      } else if (j >= 448) {
        int s = j - 448, w = s / 3, k2 = s - 3 * w;
        if (k == k2) val = lm1[u * 32 + w] * INVSQRT32;
      }
    }
    Wbig[idx] = val;
  } else if (idx < NB + 64 * 1024) {
    int t = idx - NB;
    int v = t >> 10, rest = t & 1023, u = rest >> 5, w = rest & 31;
    Wperm[t] = wu101[u * 2048 + v * 32 + w];
  }
}

// ---------------------------------------------------------------------------
// K1: F[N,544] = NF[N,160] @ Wbig[160,544]  (WMMA f32 GEMM, K=160)
// grid: (mtiles, 17), block 64 (2 waves, one 16-col tile each -> 34 tiles)
// ---------------------------------------------------------------------------
__global__ void __launch_bounds__(64)
k_precompute(const float* __restrict__ nf, const float* __restrict__ Wbig,
             float* __restrict__ F, int N) {
  __shared__ __align__(16) float snf[16 * 160];
  const int mtile = blockIdx.x;
  const int tid = threadIdx.x;
  // async-stage the 16x160 node tile (16B per lane per op)
  for (int ch = tid; ch < 640; ch += 64) {
    int f0 = ch * 4;
    int m = f0 / 160, j = f0 - m * 160;
    int node = mtile * 16 + m; if (node >= N) node = N - 1;
    async_copy16(nf + (long)node * 160 + j, &snf[f0]);
  }
  async_wait0();
  __syncthreads();
  const int lane = tid & 31;
  const int wtile = blockIdx.y * 2 + (tid >> 5);  // 0..33
  const int m = lane & 15;
  const int khi = (lane >> 4) << 1;               // 0 or 2
  const int col = wtile * 16 + (lane & 15);
  v8f acc = V8ZERO;
  for (int k0 = 0; k0 < 160; k0 += 4) {
    v2f a, b;
    a.x = snf[m * 160 + k0 + khi];
    a.y = snf[m * 160 + k0 + khi + 1];
    b.x = Wbig[(k0 + khi) * 544 + col];
    b.y = Wbig[(k0 + khi + 1) * 544 + col];
    acc = wmma_f32x4(a, b, acc);
  }
  const int hi = (lane >> 4) << 3;                // 0 or 8
  if (mtile * 16 + 16 <= N) {
#pragma unroll
    for (int g = 0; g < 8; ++g)
      F[(long)(mtile * 16 + g + hi) * 544 + col] = acc[g];
  } else {
    for (int g = 0; g < 8; ++g) {
      int node = mtile * 16 + g + hi;
      if (node < N) F[(long)node * 544 + col] = acc[g];
    }
  }
}

// ---------------------------------------------------------------------------
// K2: per-edge message from gathered F[col] (L2-resident, 109 MB < 192 MB L2)
//     + fp32 atomic scatter into agg[row].  One thread per (edge, out-chan).
// ---------------------------------------------------------------------------
__global__ void k_edge_scatter(const int* __restrict__ eidx,
                               const float* __restrict__ attr,
                               const float* __restrict__ F,
                               float* __restrict__ agg, int E) {
  unsigned gid = blockIdx.x * blockDim.x + threadIdx.x;
  unsigned total = (unsigned)E * 160u;
  if (gid >= total) return;
  int e = gid / 160u;
  int j = gid - (unsigned)e * 160u;
  int row = eidx[e];
  int col = eidx[E + e];
  const float* f = F + (long)col * 544;
  float r0 = attr[e * 4 + 0], r1 = attr[e * 4 + 1];
  float r2 = attr[e * 4 + 2], d = attr[e * 4 + 3];
  float val;
  if (j < 64) {
    int w = j;
    val = d * f[w] + r0 * f[64 + w * 3] + r1 * f[64 + w * 3 + 1] +
          r2 * f[64 + w * 3 + 2] + f[384 + w];
  } else {
    int t = j - 64, w = t / 3, k = t - 3 * w;
    float rk = (k == 0) ? r0 : ((k == 1) ? r1 : r2);
    val = rk * f[256 + w] + d * f[288 + w * 3 + k] + f[448 + w * 3 + k];
  }
  atomicAdd(&agg[(long)row * 160 + j], val);
}

// ---------------------------------------------------------------------------
// K3: u0[n,w] = ( x0 (x) a0 : wu000 + (1/sqrt3) x1 (.) a1 : wu110 ) / sqrt(5120)
// Outer-product GEMM with A built on the fly from LDS; B streamed through LDS
// with double-buffered GLOBAL_LOAD_ASYNC_TO_LDS_B128 (64-row x 64-col chunks,
// 16 KB each, contiguous in memory).
// grid: mtiles, block 128 (4 waves = 4 col-tiles of 16 -> all 64 outputs).
// ---------------------------------------------------------------------------
#define U0_CHUNK_ROWS 64
#define U0_CHUNK_FLTS (U0_CHUNK_ROWS * 64)   // 4096 floats / 16 KB
#define U0_NCHUNK_000 64                     // 4096 rows
#define U0_NCHUNK_110 16                     // 1024 rows
#define U0_NCHUNK     (U0_NCHUNK_000 + U0_NCHUNK_110)

__global__ void __launch_bounds__(128)
k_update_u0(const float* __restrict__ nf, const float* __restrict__ agg,
            const float* __restrict__ wu000, const float* __restrict__ wu110,
            float* __restrict__ u0buf, int N) {
  __shared__ __align__(16) float sx[16 * 160];
  __shared__ __align__(16) float sa[16 * 160];
  __shared__ __align__(16) float sB[2][U0_CHUNK_FLTS];
  const int mtile = blockIdx.x;
  const int tid = threadIdx.x;

  // async-stage node tiles
  for (int ch = tid; ch < 640; ch += 128) {
    int f0 = ch * 4;
    int m = f0 / 160, j = f0 - m * 160;
    int node = mtile * 16 + m; if (node >= N) node = N - 1;
    async_copy16(nf + (long)node * 160 + j, &sx[f0]);
    async_copy16(agg + (long)node * 160 + j, &sa[f0]);
  }
  // async-stage first weight chunk (contiguous 16 KB of wu000)
  for (int ch = tid; ch < U0_CHUNK_FLTS / 4; ch += 128)
    async_copy16(wu000 + ch * 4, &sB[0][ch * 4]);
  async_wait0();
  __syncthreads();

  const int lane = tid & 31;
  const int wtile = tid >> 5;                 // 0..3
  const int m = lane & 15;
  const int khi = (lane >> 4) << 1;
  const int colw = wtile * 16 + (lane & 15);
  const float* xr = sx + m * 160;
  const float* ar = sa + m * 160;
  const float* x1r = xr + 64;
  const float* a1r = ar + 64;
  v8f acc = V8ZERO;

  for (int c = 0; c < U0_NCHUNK; ++c) {
    const float* bcur = sB[c & 1];
    // issue async copy of the next weight chunk into the other buffer
    if (c + 1 < U0_NCHUNK) {
      const float* src = (c + 1 < U0_NCHUNK_000)
                             ? (wu000 + (c + 1) * U0_CHUNK_FLTS)
                             : (wu110 + (c + 1 - U0_NCHUNK_000) * U0_CHUNK_FLTS);
      float* dst = sB[(c + 1) & 1];
      for (int ch = tid; ch < U0_CHUNK_FLTS / 4; ch += 128)
        async_copy16(src + ch * 4, dst + ch * 4);
    }
    if (c < U0_NCHUNK_000) {
      // 0e x 0e -> 0e : A[n, u*64+v] = x0[n,u]*a0[n,v]
      const int base = c * U0_CHUNK_ROWS;
#pragma unroll 4
      for (int s = 0; s < U0_CHUNK_ROWS / 4; ++s) {
        int pA = base + s * 4 + khi;
        int rl = s * 4 + khi;
        v2f a, b;
        a.x = xr[pA >> 6] * ar[pA & 63];
        a.y = xr[(pA + 1) >> 6] * ar[(pA + 1) & 63];
        b.x = bcur[rl * 64 + colw];
        b.y = bcur[(rl + 1) * 64 + colw];
        acc = wmma_f32x4(a, b, acc);
      }
    } else {
      // 1o x 1o -> 0e : A[n, u*32+v] = (1/sqrt3) dot3(x1[n,u,:], a1[n,v,:])
      const int base = (c - U0_NCHUNK_000) * U0_CHUNK_ROWS;
#pragma unroll 2
      for (int s = 0; s < U0_CHUNK_ROWS / 4; ++s) {
        int pA = base + s * 4 + khi;
        int rl = s * 4 + khi;
        int uA = (pA >> 5) * 3, vA = (pA & 31) * 3;
        int uB = ((pA + 1) >> 5) * 3, vB = ((pA + 1) & 31) * 3;
        v2f a, b;
        a.x = INVSQRT3 * (x1r[uA] * a1r[vA] + x1r[uA + 1] * a1r[vA + 1] +
                          x1r[uA + 2] * a1r[vA + 2]);
        a.y = INVSQRT3 * (x1r[uB] * a1r[vB] + x1r[uB + 1] * a1r[vB + 1] +
                          x1r[uB + 2] * a1r[vB + 2]);
        b.x = bcur[rl * 64 + colw];
        b.y = bcur[(rl + 1) * 64 + colw];
        acc = wmma_f32x4(a, b, acc);
      }
    }
    async_wait0();
    __syncthreads();
  }

  const int hi = (lane >> 4) << 3;
  if (mtile * 16 + 16 <= N) {
#pragma unroll
    for (int g = 0; g < 8; ++g)
      u0buf[(long)(mtile * 16 + g + hi) * 64 + colw] = acc[g] * S5120;
  } else {
    for (int g = 0; g < 8; ++g) {
      int node = mtile * 16 + g + hi;
      if (node < N) u0buf[(long)node * 64 + colw] = acc[g] * S5120;
    }
  }
}

// ---------------------------------------------------------------------------
// K4: u1[n,w,k] = ( sum_v a1[v,k]*W2[n,v,w] + sum_u x1[u,k]*W3[n,u,w] ) / 64
//   W2[n,v,w] = sum_u x0[n,u] wu011[u,v,w]   (WMMA tile, K=64, per outer v)
//   W3[n,u,w] = sum_v a0[n,v] wu101[u,v,w]   (via Wperm, per outer u=v)
// Tiles contracted in-register against the vector irreps -> no big scratch.
// grid: mtiles, block 64 (2 waves = 2 col-tiles -> w in [0,32)).
// ---------------------------------------------------------------------------
__global__ void __launch_bounds__(64)
k_update_u1(const float* __restrict__ nf, const float* __restrict__ agg,
            const float* __restrict__ wu011, const float* __restrict__ Wperm,
            float* __restrict__ u1buf, int N) {
  __shared__ __align__(16) float sx[16 * 160];
  __shared__ __align__(16) float sa[16 * 160];
  const int mtile = blockIdx.x;
  const int tid = threadIdx.x;
  for (int ch = tid; ch < 640; ch += 64) {
    int f0 = ch * 4;
    int m = f0 / 160, j = f0 - m * 160;
    int node = mtile * 16 + m; if (node >= N) node = N - 1;
    async_copy16(nf + (long)node * 160 + j, &sx[f0]);
    async_copy16(agg + (long)node * 160 + j, &sa[f0]);
  }
  async_wait0();
  __syncthreads();
  const int lane = tid & 31;
  const int wtile = tid >> 5;                 // 0..1
  const int m = lane & 15;
  const int khi = (lane >> 4) << 1;
  const int colw = wtile * 16 + (lane & 15);
  const int hi = (lane >> 4) << 3;
  v8f acc0 = V8ZERO, acc1 = V8ZERO, acc2 = V8ZERO;
  for (int v = 0; v < 32; ++v) {
    // --- 0e x 1o path: W2 tile = x0-tile @ wu011[:, v, w-tile]
    v8f t = V8ZERO;
    const float* B = wu011 + v * 32;
    __builtin_prefetch(B + 32 + colw, 0, 0);  // next v-slab
#pragma unroll 4
    for (int k0 = 0; k0 < 64; k0 += 4) {
      int r = k0 + khi;
      v2f a, b;
      a.x = sx[m * 160 + r];
      a.y = sx[m * 160 + r + 1];
      b.x = B[r * 1024 + colw];
      b.y = B[(r + 1) * 1024 + colw];
      t = wmma_f32x4(a, b, t);
    }
#pragma unroll
    for (int g = 0; g < 8; ++g) {
      const float* a1n = sa + (g + hi) * 160 + 64 + v * 3;
      float wv = t[g];
      acc0[g] += a1n[0] * wv;
      acc1[g] += a1n[1] * wv;
      acc2[g] += a1n[2] * wv;
    }
    // --- 1o x 0e path: W3 tile = a0-tile @ Wperm[:, u=v, w-tile]
    t = V8ZERO;
    const float* B3 = Wperm + v * 32;
#pragma unroll 4
    for (int k0 = 0; k0 < 64; k0 += 4) {
      int r = k0 + khi;
      v2f a, b;
      a.x = sa[m * 160 + r];
      a.y = sa[m * 160 + r + 1];
      b.x = B3[r * 1024 + colw];
      b.y = B3[(r + 1) * 1024 + colw];
      t = wmma_f32x4(a, b, t);
    }
#pragma unroll
    for (int g = 0; g < 8; ++g) {
      const float* x1n = sx + (g + hi) * 160 + 64 + v * 3;
      float wv = t[g];
      acc0[g] += x1n[0] * wv;
      acc1[g] += x1n[1] * wv;
      acc2[g] += x1n[2] * wv;
    }
  }
  for (int g = 0; g < 8; ++g) {
    int node = mtile * 16 + g + hi;
    if (node < N) {
      float* o = u1buf + (long)node * 96 + colw * 3;
      o[0] = acc0[g] * S4096;
      o[1] = acc1[g] * S4096;
      o[2] = acc2[g] * S4096;
    }
  }
}

// ---------------------------------------------------------------------------
// K5: out = nf + u0 + x0@lu0/8  |  u1 + x1@lu1/sqrt(32)   (residual + linears)
// ---------------------------------------------------------------------------
__global__ void k_output(const float* __restrict__ nf,
                         const float* __restrict__ u0buf,
                         const float* __restrict__ u1buf,
                         const float* __restrict__ lu0,
                         const float* __restrict__ lu1,
                         float* __restrict__ out, int N) {
  int gid = blockIdx.x * blockDim.x + threadIdx.x;
  int total = N * 160;
  if (gid >= total) return;
  int n = gid / 160;
  int j = gid - n * 160;
  const float* x = nf + (long)n * 160;
  float res = x[j];
  if (j < 64) {
    res += u0buf[(long)n * 64 + j];
    float s = 0.f;
#pragma unroll 8
    for (int u = 0; u < 64; ++u) s += x[u] * lu0[u * 64 + j];
    res += s * 0.125f;
  } else {
    int t = j - 64, w = t / 3, k = t - 3 * w;
    res += u1buf[(long)n * 96 + w * 3 + k];
    float s = 0.f;
#pragma unroll 8
    for (int u = 0; u < 32; ++u) s += x[64 + u * 3 + k] * lu1[u * 32 + w];
    res += s * INVSQRT32;
  }
  out[gid] = res;
}

// ---------------------------------------------------------------------------
extern "C" void kernel_launch(void* const* d_in, const int* in_sizes, int n_in,
                              void* d_out, int out_size, void* d_ws,
                              size_t ws_size, hipStream_t stream) {
  (void)n_in; (void)out_size; (void)ws_size;
  const float* nf    = (const float*)d_in[0];
  const int*   eidx  = (const int*)d_in[1];
  const float* attr  = (const float*)d_in[2];
  // d_in[3] node_attr_scalar_raw: unused by the reference
  const float* wm000 = (const float*)d_in[4];
  const float* wm011 = (const float*)d_in[5];
  const float* wm101 = (const float*)d_in[6];
  const float* wm110 = (const float*)d_in[7];
  const float* lm0   = (const float*)d_in[8];
  const float* lm1   = (const float*)d_in[9];
  const float* wu000 = (const float*)d_in[10];
  const float* wu011 = (const float*)d_in[11];
  const float* wu101 = (const float*)d_in[12];
  const float* wu110 = (const float*)d_in[13];
  const float* lu0   = (const float*)d_in[14];
  const float* lu1   = (const float*)d_in[15];

  const int N = in_sizes[0] / 160;
  const int E = in_sizes[1] / 2;

  char* ws = (char*)d_ws;
  size_t off = 0;
  auto alloc = [&](size_t bytes) -> void* {
    void* p = ws + off;
    off = (off + bytes + 255) & ~(size_t)255;
    return p;
  };
  float* Wbig  = (float*)alloc((size_t)160 * 544 * 4);
  float* Wperm = (float*)alloc((size_t)64 * 1024 * 4);
  float* F     = (float*)alloc((size_t)N * 544 * 4);   // ~109 MB (L2-resident)
  float* agg   = (float*)alloc((size_t)N * 160 * 4);   // 32 MB
  float* u0buf = (float*)alloc((size_t)N * 64 * 4);
  float* u1buf = (float*)alloc((size_t)N * 96 * 4);

  (void)hipMemsetAsync(agg, 0, (size_t)N * 160 * 4, stream);

  const int totW = 160 * 544 + 64 * 1024;
  k_build_weights<<<(totW + 255) / 256, 256, 0, stream>>>(
      wm000, wm011, wm101, wm110, lm0, lm1, wu101, Wbig, Wperm);

  const int mtiles = (N + 15) / 16;
  k_precompute<<<dim3(mtiles, 17), 64, 0, stream>>>(nf, Wbig, F, N);

  const unsigned totE = (unsigned)E * 160u;
  k_edge_scatter<<<(totE + 255u) / 256u, 256, 0, stream>>>(eidx, attr, F, agg, E);

  k_update_u0<<<mtiles, 128, 0, stream>>>(nf, agg, wu000, wu110, u0buf, N);
  k_update_u1<<<mtiles, 64, 0, stream>>>(nf, agg, wu011, Wperm, u1buf, N);

  k_output<<<(N * 160 + 255) / 256, 256, 0, stream>>>(nf, u0buf, u1buf, lu0,
                                                      lu1, (float*)d_out, N);
}